// TestNet3_24257975287987
// MI455X (gfx1250) — compile-verified
//
#include <hip/hip_runtime.h>

#define EPS 1e-5f
#define SLOPE 0.01f

typedef float v2f __attribute__((ext_vector_type(2)));
typedef float v8f __attribute__((ext_vector_type(8)));
typedef int   v4i __attribute__((ext_vector_type(4)));

#if __has_builtin(__builtin_amdgcn_global_load_async_to_lds_b128) && \
    __has_builtin(__builtin_amdgcn_s_wait_asynccnt)
#define USE_ASYNC_LDS 1
typedef __attribute__((address_space(1))) v4i* glob_v4i_ptr;
typedef __attribute__((address_space(3))) v4i* lds_v4i_ptr;
#else
#define USE_ASYNC_LDS 0
#endif

// ---------------------------------------------------------------------------
// Float atomic max via sign-split int atomics (init with -inf).
// ---------------------------------------------------------------------------
__device__ __forceinline__ void atomicMaxFloat(float* addr, float val) {
    if (val >= 0.0f) {
        atomicMax((int*)addr, __float_as_int(val));
    } else {
        atomicMin((unsigned int*)addr, __float_as_uint(val));
    }
}

__global__ void k_fill(float* __restrict__ p, float v, int n) {
    int i = blockIdx.x * blockDim.x + threadIdx.x;
    if (i < n) p[i] = v;
}

__global__ void k_degree(const long long* __restrict__ dst, float* __restrict__ deg, int E) {
    int i = blockIdx.x * blockDim.x + threadIdx.x;
    if (i < E) atomicAdd(&deg[(int)dst[i]], 1.0f);
}

__global__ void k_dinv(const float* __restrict__ deg, float* __restrict__ dinv, int n) {
    int i = blockIdx.x * blockDim.x + threadIdx.x;
    if (i < n) dinv[i] = rsqrtf(deg[i]);
}

// y[n][c] = dinv[n]^2 * x[n][c]  (self-loop term)
__global__ void k_selfloop(const float* __restrict__ x, const float* __restrict__ dinv,
                           float* __restrict__ y, int total, int C) {
    int i = blockIdx.x * blockDim.x + threadIdx.x;
    if (i < total) {
        float d = dinv[i / C];
        y[i] = d * d * x[i];
    }
}

// Scalar edge scatter (layer 1, C=3)
__global__ void k_edge_agg(const long long* __restrict__ src, const long long* __restrict__ dst,
                           const float* __restrict__ dinv, const float* __restrict__ x,
                           float* __restrict__ y, int total, int C) {
    int i = blockIdx.x * blockDim.x + threadIdx.x;
    if (i < total) {
        int e = i / C, c = i % C;
        int s = (int)src[e], d = (int)dst[e];
        float w = dinv[s] * dinv[d];
        atomicAdd(&y[d * C + c], w * x[s * C + c]);
    }
}

// Vectorized edge scatter: one thread = 4 channels of one edge (C % 4 == 0)
__global__ void k_edge_agg4(const long long* __restrict__ src, const long long* __restrict__ dst,
                            const float* __restrict__ dinv, const float* __restrict__ x,
                            float* __restrict__ y, int total4, int C) {
    int i = blockIdx.x * blockDim.x + threadIdx.x;
    if (i < total4) {
        const int cq = C >> 2;
        int e = i / cq, c = (i % cq) << 2;
        int s = (int)src[e], d = (int)dst[e];
        float w = dinv[s] * dinv[d];
        float4 xv = *(const float4*)(x + s * C + c);
        float* yp = y + d * C + c;
        atomicAdd(yp + 0, w * xv.x);
        atomicAdd(yp + 1, w * xv.y);
        atomicAdd(yp + 2, w * xv.z);
        atomicAdd(yp + 3, w * xv.w);
    }
}

// Layer-1 GEMM (Cin=3): trivial FLOPs, keep it branch-free scalar.
__global__ void k_gemm3(const float* __restrict__ Y, const float* __restrict__ W,
                        const float* __restrict__ bias,
                        const float* __restrict__ bg, const float* __restrict__ bb,
                        const float* __restrict__ bm, const float* __restrict__ bv,
                        float* __restrict__ out, int total, int Cout) {
    int i = blockIdx.x * blockDim.x + threadIdx.x;
    if (i >= total) return;
    int n = i / Cout, j = i % Cout;
    float acc = bias[j] + Y[n * 3 + 0] * W[j] + Y[n * 3 + 1] * W[Cout + j] +
                Y[n * 3 + 2] * W[2 * Cout + j];
    float sc = bg[j] * rsqrtf(bv[j] + EPS);
    float t = (acc - bm[j]) * sc + bb[j];
    out[i] = (t >= 0.0f) ? t : SLOPE * t;
}

// ---------------------------------------------------------------------------
// WMMA GEMM: D = Y(N x CIN) * W(CIN x Cout) fused +bias +BN +LReLU [+max-pool].
// Block = 128 threads (4 waves) -> 64 output columns; B tile (CIN x 64) staged
// once into LDS (async global->LDS on CDNA5), then 5 row-tiles of 16 nodes.
// A frag (16x4 f32): lanes 0-15 K={k0,k0+1}, lanes 16-31 K={k0+2,k0+3}.
// C/D: VGPR r -> (M = r + 8*khalf, N = lane&15).
// ---------------------------------------------------------------------------
template <int CIN, bool POOL>
__global__ void k_gemm_wmma(const float* __restrict__ Y, const float* __restrict__ W,
                            const float* __restrict__ bias,
                            const float* __restrict__ bg, const float* __restrict__ bb,
                            const float* __restrict__ bm, const float* __restrict__ bv,
                            float* __restrict__ out, int Cout, int tilesPerBlock) {
    constexpr int PAD = 68;  // 64 + 4: conflict-free column reads, 16B-aligned rows
    __shared__ __align__(16) float Blds[CIN * PAD];

    const int tid = threadIdx.x;
    const int colBlock = blockIdx.y * 64;

    // ---- Stage B tile (CIN x 64) into LDS ----
#if USE_ASYNC_LDS
    for (int q = tid; q < CIN * 16; q += 128) {
        const int r = q >> 4, c4 = (q & 15) << 2;
        __builtin_amdgcn_global_load_async_to_lds_b128(
            (glob_v4i_ptr)(W + r * Cout + colBlock + c4),
            (lds_v4i_ptr)(&Blds[r * PAD + c4]), 0, 0);
    }
    __builtin_amdgcn_s_wait_asynccnt(0);
#else
    for (int q = tid; q < CIN * 16; q += 128) {
        const int r = q >> 4, c4 = (q & 15) << 2;
        *(float4*)&Blds[r * PAD + c4] = *(const float4*)(W + r * Cout + colBlock + c4);
    }
#endif
    __syncthreads();

    const int wave  = tid >> 5;
    const int lane  = tid & 31;
    const int mrow  = lane & 15;
    const int khalf = lane >> 4;
    const int colL  = wave * 16 + mrow;  // 0..63
    const int col   = colBlock + colL;

    const float bi = bias[col];
    const float sc = bg[col] * rsqrtf(bv[col] + EPS);
    const float sh = bb[col] - bm[col] * sc;

    float mx = -__builtin_inff();

    for (int t = 0; t < tilesPerBlock; ++t) {
        const int row_base = (blockIdx.x * tilesPerBlock + t) * 16;
        const float* Yrow = Y + (row_base + mrow) * CIN;

        v8f acc = {0.f, 0.f, 0.f, 0.f, 0.f, 0.f, 0.f, 0.f};
#pragma unroll
        for (int k0 = 0; k0 < CIN; k0 += 4) {
            const int ka = k0 + 2 * khalf;
            v2f a = *(const v2f*)(Yrow + ka);           // aligned b64
            v2f b;
            b.x = Blds[ka * PAD + colL];
            b.y = Blds[(ka + 1) * PAD + colL];
            acc = __builtin_amdgcn_wmma_f32_16x16x4_f32(false, a, false, b,
                                                        (short)0, acc, false, false);
        }

        if (POOL) {
#pragma unroll
            for (int r = 0; r < 8; ++r) {
                float v = (acc[r] + bi) * sc + sh;
                v = (v >= 0.0f) ? v : SLOPE * v;
                mx = fmaxf(mx, v);
            }
        } else {
#pragma unroll
            for (int r = 0; r < 8; ++r) {
                float v = (acc[r] + bi) * sc + sh;
                v = (v >= 0.0f) ? v : SLOPE * v;
                out[(row_base + r + khalf * 8) * Cout + col] = v;
            }
        }
    }
    if (POOL) atomicMaxFloat(&out[col], mx);
}

// Small FC: one thread per output column (coalesced W columns).
__global__ void k_fc_bn(const float* __restrict__ xin, const float* __restrict__ W,
                        const float* __restrict__ bias,
                        const float* __restrict__ g, const float* __restrict__ b,
                        const float* __restrict__ m, const float* __restrict__ v,
                        float* __restrict__ xout, int K, int Cout, int do_lrelu) {
    int j = blockIdx.x * blockDim.x + threadIdx.x;
    if (j >= Cout) return;
    float acc = bias[j];
    for (int k = 0; k < K; ++k) acc += xin[k] * W[k * Cout + j];
    float t = (acc - m[j]) * rsqrtf(v[j] + EPS) * g[j] + b[j];
    if (do_lrelu) t = (t >= 0.0f) ? t : SLOPE * t;
    xout[j] = t;
}

extern "C" void kernel_launch(void* const* d_in, const int* in_sizes, int n_in,
                              void* d_out, int out_size, void* d_ws, size_t ws_size,
                              hipStream_t stream) {
    (void)n_in; (void)out_size; (void)ws_size;

    const float*     pos  = (const float*)d_in[0];
    const long long* eidx = (const long long*)d_in[1];
    const int N = in_sizes[0] / 3;      // 10000
    const int E = in_sizes[1] / 2;      // 160000
    const long long* src = eidx;
    const long long* dst = eidx + E;

    const float *convW[5], *convB[5], *bnG[5], *bnB[5], *bnM[5], *bnV[5];
    for (int l = 0; l < 5; ++l) {
        convW[l] = (const float*)d_in[2 + l];
        convB[l] = (const float*)d_in[7 + l];
        bnG[l]   = (const float*)d_in[12 + l];
        bnB[l]   = (const float*)d_in[17 + l];
        bnM[l]   = (const float*)d_in[22 + l];
        bnV[l]   = (const float*)d_in[27 + l];
    }
    const float* W1  = (const float*)d_in[32];
    const float* b1  = (const float*)d_in[33];
    const float* g1  = (const float*)d_in[34];
    const float* be1 = (const float*)d_in[35];
    const float* m1  = (const float*)d_in[36];
    const float* v1  = (const float*)d_in[37];
    const float* W2  = (const float*)d_in[38];
    const float* b2  = (const float*)d_in[39];
    const float* g2  = (const float*)d_in[40];
    const float* be2 = (const float*)d_in[41];
    const float* m2  = (const float*)d_in[42];
    const float* v2  = (const float*)d_in[43];

    // ws layout (floats): deg | dinv | bufA(N*128) | bufB(N*128) | pooled(2048) | h1(512)
    float* ws     = (float*)d_ws;
    float* deg    = ws;
    float* dinv   = ws + N;
    float* bufA   = dinv + N;
    float* bufB   = bufA + (size_t)N * 128;
    float* pooled = bufB + (size_t)N * 128;
    float* h1     = pooled + 2048;

    const int T = 256;
    const int TILES = 5;                 // row tiles per block
    const int gx = N / 16 / TILES;       // 125

    // Degree (incl. self loop) and D^{-1/2}
    k_fill<<<(N + T - 1) / T, T, 0, stream>>>(deg, 1.0f, N);
    k_degree<<<(E + T - 1) / T, T, 0, stream>>>(dst, deg, E);
    k_dinv<<<(N + T - 1) / T, T, 0, stream>>>(deg, dinv, N);

    // ---- Layer 1: aggregate(pos)[N x 3] -> bufA ; GEMM 3->64 -> bufB ----
    k_selfloop<<<(N * 3 + T - 1) / T, T, 0, stream>>>(pos, dinv, bufA, N * 3, 3);
    k_edge_agg<<<(E * 3 + T - 1) / T, T, 0, stream>>>(src, dst, dinv, pos, bufA, E * 3, 3);
    k_gemm3<<<(N * 64 + T - 1) / T, T, 0, stream>>>(bufA, convW[0], convB[0],
        bnG[0], bnB[0], bnM[0], bnV[0], bufB, N * 64, 64);

    // ---- Layers 2..5 (WMMA path, reordered GCN: (A^ x) W) ----
    const int chs[6] = {3, 64, 64, 64, 128, 2048};
    for (int l = 1; l < 5; ++l) {
        const int Cin = chs[l], Cout = chs[l + 1];
        k_selfloop<<<(N * Cin + T - 1) / T, T, 0, stream>>>(bufB, dinv, bufA, N * Cin, Cin);
        const int tot4 = E * (Cin / 4);
        k_edge_agg4<<<(tot4 + T - 1) / T, T, 0, stream>>>(src, dst, dinv, bufB, bufA, tot4, Cin);

        dim3 grid(gx, Cout / 64);
        if (l < 4) {
            k_gemm_wmma<64, false><<<grid, 128, 0, stream>>>(
                bufA, convW[l], convB[l], bnG[l], bnB[l], bnM[l], bnV[l], bufB, Cout, TILES);
        } else {
            k_fill<<<(2048 + T - 1) / T, T, 0, stream>>>(pooled, -__builtin_inff(), 2048);
            k_gemm_wmma<128, true><<<grid, 128, 0, stream>>>(
                bufA, convW[l], convB[l], bnG[l], bnB[l], bnM[l], bnV[l], pooled, Cout, TILES);
        }
    }

    // FC1: 2048 -> 512 (+BN +LReLU), FC2: 512 -> 4096 (+BN)
    k_fc_bn<<<(512 + T - 1) / T, T, 0, stream>>>(pooled, W1, b1, g1, be1, m1, v1, h1, 2048, 512, 1);
    k_fc_bn<<<(4096 + T - 1) / T, T, 0, stream>>>(h1, W2, b2, g2, be2, m2, v2, (float*)d_out, 512, 4096, 0);
}